// HardGatedExperts_64029372448801
// MI455X (gfx1250) — compile-verified
//
#include <hip/hip_runtime.h>

// Problem constants (match reference)
#define ED   512      // D
#define EH1  512
#define EH2  256
#define EY   256
#define NEXP 8
#define LALPHA 0.2f
#define TM   32       // token rows per workgroup tile
#define AS   (ED + 4)   // LDS row stride for A / ACT1 (516: 4m+k bank spread, 8B aligned)
#define A2S  (EH2 + 4)  // LDS row stride for ACT2 (260)

typedef float v2f __attribute__((ext_vector_type(2)));
typedef float v8f __attribute__((ext_vector_type(8)));

// ---------------- routing pass ----------------

__global__ void k_zero(int* ws) {
    if (threadIdx.x < 26) ws[threadIdx.x] = 0;   // counts[8], offsets[9], cursor[8], flag[1]
}

// flag=1  -> gate buffer is int32; flag=0 -> int64 (odd 32-bit words all zero)
__global__ void k_detect(const int* g32, int* flag, int B) {
    int i = blockIdx.x * blockDim.x + threadIdx.x;
    if (i < B && (i & 1) && g32[i] != 0) atomicOr(flag, 1);
}

__device__ __forceinline__ int load_gate(const void* gate, int i, int is32) {
    int e = is32 ? ((const int*)gate)[i] : (int)((const long long*)gate)[i];
    return min(max(e, 0), NEXP - 1);
}

__global__ void k_count(const void* gate, const int* flag, int* counts, int B) {
    int i = blockIdx.x * blockDim.x + threadIdx.x;
    if (i < B) atomicAdd(&counts[load_gate(gate, i, *flag)], 1);
}

__global__ void k_scan(const int* counts, int* offsets, int* cursor) {
    if (threadIdx.x == 0) {
        int acc = 0;
        for (int e = 0; e < NEXP; e++) { offsets[e] = acc; cursor[e] = acc; acc += counts[e]; }
        offsets[NEXP] = acc;
    }
}

__global__ void k_fill(const void* gate, const int* flag, int* cursor, int* perm, int B) {
    int i = blockIdx.x * blockDim.x + threadIdx.x;
    if (i < B) {
        int e = load_gate(gate, i, *flag);
        int slot = atomicAdd(&cursor[e], 1);
        perm[slot] = i;
    }
}

// ---------------- expert MLP (WMMA f32 16x16x4) ----------------

__device__ __forceinline__ float leaky(float x) { return x > 0.f ? x : LALPHA * x; }

// One 32xN(=16) output tile: D = A(32xK from LDS) * W(KxN global) , two 16-row WMMA subtiles.
// f32 16x16x4 layouts: A lane l, vgpr v: M=l%16, K=2*(l/16)+v.  B mirrored: K=2*(l/16)+v, N=l%16.
template <int K, int N, int LDA>
__device__ __forceinline__ void gemm_tile(const float* __restrict__ Asrc,
                                          const float* __restrict__ Wg,
                                          int nb, int lane, v8f& acc0, v8f& acc1) {
    const int hi = lane >> 4;
    const int m  = lane & 15;
    const int n  = nb + m;
    const float* __restrict__ w0 = Wg + (2 * hi) * N + n;     // + kb*N, +N for second k-row
    const float* __restrict__ a0 = Asrc + m * LDA + 2 * hi;   // + kb
    const float* __restrict__ a1 = Asrc + (m + 16) * LDA + 2 * hi;
    acc0 = (v8f){}; acc1 = (v8f){};
#pragma unroll 8
    for (int kb = 0; kb < K; kb += 4) {
        v2f b;
        b.x = w0[kb * N];
        b.y = w0[kb * N + N];
        v2f av0 = *(const v2f*)(a0 + kb);
        v2f av1 = *(const v2f*)(a1 + kb);
        acc0 = __builtin_amdgcn_wmma_f32_16x16x4_f32(false, av0, false, b, (short)0, acc0, false, false);
        acc1 = __builtin_amdgcn_wmma_f32_16x16x4_f32(false, av1, false, b, (short)0, acc1, false, false);
    }
}

__launch_bounds__(256, 1)
__global__ void k_moe(const float* __restrict__ hin, const int* __restrict__ perm,
                      const int* __restrict__ offsets,
                      const float* __restrict__ W1, const float* __restrict__ b1,
                      const float* __restrict__ W2, const float* __restrict__ b2,
                      const float* __restrict__ W3, const float* __restrict__ b3,
                      float* __restrict__ out) {
    extern __shared__ float smem[];
    float* A    = smem;             // TM x AS   (h tile; reused as ACT2 after layer1)
    float* ACT1 = smem + TM * AS;   // TM x AS
    float* ACT2 = smem;             // alias of A (TM x A2S <= TM x AS)
    __shared__ int tokens[TM];

    const int tid  = threadIdx.x;
    const int lane = tid & 31;
    const int wave = tid >> 5;

    // Locate this block's (expert, row range) from tile id.
    int t = blockIdx.x;
    int e = -1, rowStart = 0, rowEnd = 0;
    int off0 = offsets[0];
    for (int i = 0; i < NEXP; i++) {
        int off1 = offsets[i + 1];
        int nt = (off1 - off0 + TM - 1) / TM;
        if (e < 0) {
            if (t < nt) { e = i; rowStart = off0 + t * TM; rowEnd = off1; }
            else t -= nt;
        }
        off0 = off1;
    }
    if (e < 0) return;

    // ---- stage 32 gathered token rows into LDS ----
    if (tid < TM) {
        int g = rowStart + tid;
        tokens[tid] = perm[g < rowEnd ? g : rowStart];
    }
    {
        int r  = tid >> 3;            // 8 threads per row
        int c0 = (tid & 7) * 64;      // 64 floats each
        int g  = rowStart + r;
        int tok = perm[g < rowEnd ? g : rowStart];
        const float4* src = (const float4*)(hin + (size_t)tok * ED + c0);
        float4* dst = (float4*)(A + r * AS + c0);
#pragma unroll
        for (int j = 0; j < 16; j++) dst[j] = src[j];
    }
    __syncthreads();

    // ---- layer 1: [32x512] x [512x512] -> ACT1, leaky ----
    const float* W1e = W1 + (size_t)e * ED * EH1;
    const float* b1e = b1 + (size_t)e * EH1;
    for (int i = 0; i < 4; i++) {
        int nb = (wave * 4 + i) * 16;
        v8f c0, c1;
        gemm_tile<ED, EH1, AS>(A, W1e, nb, lane, c0, c1);
        int n = nb + (lane & 15);
        int hi = lane >> 4;
        float bias = b1e[n];
#pragma unroll
        for (int j = 0; j < 8; j++) {
            ACT1[(j + 8 * hi) * AS + n]      = leaky(c0[j] + bias);
            ACT1[(16 + j + 8 * hi) * AS + n] = leaky(c1[j] + bias);
        }
    }
    __syncthreads();   // ACT1 complete; A no longer needed -> reuse as ACT2

    // ---- layer 2: [32x512] x [512x256] -> ACT2, leaky ----
    const float* W2e = W2 + (size_t)e * EH1 * EH2;
    const float* b2e = b2 + (size_t)e * EH2;
    for (int i = 0; i < 2; i++) {
        int nb = (wave * 2 + i) * 16;
        v8f c0, c1;
        gemm_tile<EH1, EH2, AS>(ACT1, W2e, nb, lane, c0, c1);
        int n = nb + (lane & 15);
        int hi = lane >> 4;
        float bias = b2e[n];
#pragma unroll
        for (int j = 0; j < 8; j++) {
            ACT2[(j + 8 * hi) * A2S + n]      = leaky(c0[j] + bias);
            ACT2[(16 + j + 8 * hi) * A2S + n] = leaky(c1[j] + bias);
        }
    }
    __syncthreads();

    // ---- layer 3: [32x256] x [256x256] -> out (scatter by token) ----
    const float* W3e = W3 + (size_t)e * EH2 * EY;
    const float* b3e = b3 + (size_t)e * EY;
    for (int i = 0; i < 2; i++) {
        int nb = (wave * 2 + i) * 16;
        v8f c0, c1;
        gemm_tile<EH2, EY, A2S>(ACT2, W3e, nb, lane, c0, c1);
        int n = nb + (lane & 15);
        int hi = lane >> 4;
        float bias = b3e[n];
#pragma unroll
        for (int j = 0; j < 8; j++) {
            int m0 = j + 8 * hi;
            if (rowStart + m0 < rowEnd)
                out[(size_t)tokens[m0] * EY + n] = c0[j] + bias;
            int m1 = 16 + j + 8 * hi;
            if (rowStart + m1 < rowEnd)
                out[(size_t)tokens[m1] * EY + n] = c1[j] + bias;
        }
    }
}

// ---------------- host ----------------

extern "C" void kernel_launch(void* const* d_in, const int* in_sizes, int n_in,
                              void* d_out, int out_size, void* d_ws, size_t ws_size,
                              hipStream_t stream) {
    const float* h    = (const float*)d_in[0];
    const void*  gate = d_in[1];                 // int64 or int32, auto-detected
    const float* W1   = (const float*)d_in[2];
    const float* b1   = (const float*)d_in[3];
    const float* W2   = (const float*)d_in[4];
    const float* b2   = (const float*)d_in[5];
    const float* W3   = (const float*)d_in[6];
    const float* b3   = (const float*)d_in[7];
    float* out = (float*)d_out;

    const int B = in_sizes[0] / ED;

    int* ws      = (int*)d_ws;
    int* counts  = ws;          // 8
    int* offsets = ws + 8;      // 9
    int* cursor  = ws + 17;     // 8
    int* flag    = ws + 25;     // 1
    int* perm    = ws + 26;     // B

    const int thr = 256;
    const int blk = (B + thr - 1) / thr;

    k_zero<<<1, 32, 0, stream>>>(ws);
    k_detect<<<blk, thr, 0, stream>>>((const int*)gate, flag, B);
    k_count<<<blk, thr, 0, stream>>>(gate, flag, counts, B);
    k_scan<<<1, 1, 0, stream>>>(counts, offsets, cursor);
    k_fill<<<blk, thr, 0, stream>>>(gate, flag, cursor, perm, B);

    const int maxTiles = (B + TM - 1) / TM + NEXP;          // worst-case per-expert padding
    const size_t smemBytes = (size_t)(2 * TM * AS) * sizeof(float);  // ~129 KB
    k_moe<<<maxTiles, 256, smemBytes, stream>>>(h, perm, offsets,
                                                W1, b1, W2, b2, W3, b3, out);
}